// JobEmbedding_4776003633687
// MI455X (gfx1250) — compile-verified
//
#include <hip/hip_runtime.h>
#include <hip/hip_bf16.h>
#include <math.h>

typedef float v2f __attribute__((ext_vector_type(2)));
typedef float v8f __attribute__((ext_vector_type(8)));

#define NJ      100000
#define D_IN    128
#define OUTD    256
#define HEADS   4
#define CH      64
#define E_EDGES 250000
#define NEG_SLOPE 0.2f
#define LN_EPS  1e-5f

// ---------- ordered-uint encoding for float atomicMax ----------
__device__ __forceinline__ unsigned int f2ord(float f) {
  unsigned int u = __float_as_uint(f);
  return (u & 0x80000000u) ? ~u : (u | 0x80000000u);
}
__device__ __forceinline__ float ord2f(unsigned int k) {
  return (k & 0x80000000u) ? __uint_as_float(k & 0x7FFFFFFFu)
                           : __uint_as_float(~k);
}

// ---------- fold att vectors through W: wfold[r][k][h] = sum_c W[r][k][h*64+c]*att[r][h][c]
__global__ void fold_att_kernel(const float* __restrict__ Ws,
                                const float* __restrict__ att_src,
                                const float* __restrict__ att_dst,
                                float* __restrict__ wfold_s,
                                float* __restrict__ wfold_d) {
  int idx = blockIdx.x * blockDim.x + threadIdx.x; // r*512 + k*4 + h
  if (idx >= 5 * D_IN * HEADS) return;
  int h = idx & 3;
  int k = (idx >> 2) & 127;
  int r = idx >> 9;
  const float* W  = Ws + ((size_t)r * D_IN + k) * OUTD + h * CH;
  const float* as = att_src + ((size_t)r * HEADS + h) * CH;
  const float* ad = att_dst + ((size_t)r * HEADS + h) * CH;
  float ss = 0.f, sd = 0.f;
  for (int c = 0; c < CH; ++c) { float w = W[c]; ss += w * as[c]; sd += w * ad[c]; }
  wfold_s[idx] = ss;
  wfold_d[idx] = sd;
}

// ---------- per-node attention scalars: out[n][h] = x[n,:] @ wfold[:,h]  (wave per node)
__global__ void att_proj_kernel(const float* __restrict__ x, int n_nodes,
                                const float* __restrict__ wfold, // [128][4]
                                float* __restrict__ out) {       // [n][4]
  int wave = (int)((blockIdx.x * blockDim.x + threadIdx.x) >> 5);
  int lane = threadIdx.x & 31;
  if (wave >= n_nodes) return;
  float4 xv = ((const float4*)(x + (size_t)wave * D_IN))[lane];
  float xs[4] = {xv.x, xv.y, xv.z, xv.w};
  const float* wf = wfold + lane * 16; // k = lane*4 + j, layout [k][h]
  float p0 = 0.f, p1 = 0.f, p2 = 0.f, p3 = 0.f;
#pragma unroll
  for (int j = 0; j < 4; ++j) {
    p0 += xs[j] * wf[j * 4 + 0];
    p1 += xs[j] * wf[j * 4 + 1];
    p2 += xs[j] * wf[j * 4 + 2];
    p3 += xs[j] * wf[j * 4 + 3];
  }
#pragma unroll
  for (int off = 16; off >= 1; off >>= 1) {
    p0 += __shfl_xor(p0, off, 32);
    p1 += __shfl_xor(p1, off, 32);
    p2 += __shfl_xor(p2, off, 32);
    p3 += __shfl_xor(p3, off, 32);
  }
  if (lane == 0) {
    *(float4*)(out + (size_t)wave * 4) = make_float4(p0, p1, p2, p3);
  }
}

// ---------- fp32 WMMA 16x16 tile, K=128 (32 x V_WMMA_F32_16X16X4_F32) ----------
__device__ __forceinline__ v8f wmma_tile_f32(const float* __restrict__ X,
                                             const float* __restrict__ W,
                                             int row0, int nrows, int col0) {
  int lane = threadIdx.x & 31;
  int half = lane >> 4;
  int mr   = lane & 15;
  int arow = row0 + mr; if (arow >= nrows) arow = nrows - 1; // clamp (dup rows, stores guarded)
  const float* ap = X + (size_t)arow * D_IN;
  const float* bp = W + col0 + mr;
  v8f c = {0.f, 0.f, 0.f, 0.f, 0.f, 0.f, 0.f, 0.f};
#pragma unroll 4
  for (int k0 = 0; k0 < D_IN; k0 += 4) {
    int ka = k0 + half * 2;
    v2f a; a.x = ap[ka];                  a.y = ap[ka + 1];
    v2f b; b.x = bp[(size_t)ka * OUTD];   b.y = bp[(size_t)(ka + 1) * OUTD];
    c = __builtin_amdgcn_wmma_f32_16x16x4_f32(false, a, false, b, (short)0, c, false, false);
  }
  return c;
}

// hs = X @ W  ([nrows,128] x [128,256]); one wave per 16x16 output tile
__global__ void gemm_hs_kernel(const float* __restrict__ X, const float* __restrict__ W,
                               float* __restrict__ Y, int nrows) {
  int wave = (int)((blockIdx.x * blockDim.x + threadIdx.x) >> 5);
  int lane = threadIdx.x & 31;
  int ntm  = (nrows + 15) >> 4;
  int tm = wave >> 4;   // 16 column tiles (256/16)
  int tn = wave & 15;
  if (tm >= ntm) return;                       // wave-uniform: EXEC stays all-1 for WMMA
  v8f c = wmma_tile_f32(X, W, tm * 16, nrows, tn * 16);
  int mbase = tm * 16 + (lane >> 4) * 8;
  int col   = tn * 16 + (lane & 15);
#pragma unroll
  for (int j = 0; j < 8; ++j) {
    int row = mbase + j;
    if (row < nrows) Y[(size_t)row * OUTD + col] = c[j];
  }
}

// d_out init = x_job @ W_res + sum_r biases[r]
__global__ void gemm_res_init_kernel(const float* __restrict__ X, const float* __restrict__ W,
                                     const float* __restrict__ biases,
                                     float* __restrict__ Y, int nrows) {
  int wave = (int)((blockIdx.x * blockDim.x + threadIdx.x) >> 5);
  int lane = threadIdx.x & 31;
  int ntm  = (nrows + 15) >> 4;
  int tm = wave >> 4;
  int tn = wave & 15;
  if (tm >= ntm) return;
  v8f c = wmma_tile_f32(X, W, tm * 16, nrows, tn * 16);
  int mbase = tm * 16 + (lane >> 4) * 8;
  int col   = tn * 16 + (lane & 15);
  float bsum = biases[col] + biases[OUTD + col] + biases[2 * OUTD + col] +
               biases[3 * OUTD + col] + biases[4 * OUTD + col];
#pragma unroll
  for (int j = 0; j < 8; ++j) {
    int row = mbase + j;
    if (row < nrows) Y[(size_t)row * OUTD + col] = c[j] + bsum;
  }
}

// ---------- fills ----------
__global__ void fill_u32_kernel(unsigned int* __restrict__ p, unsigned int val, int n) {
  int i = blockIdx.x * blockDim.x + threadIdx.x;
  if (i < n) p[i] = val;
}

// ---------- edge pass 1: segment max (ordered-uint atomicMax) ----------
__global__ void edge_max_kernel(const int* __restrict__ src, const int* __restrict__ dst,
                                const float* __restrict__ a_s, const float* __restrict__ a_d,
                                unsigned int* __restrict__ mkey) {
  int i = blockIdx.x * blockDim.x + threadIdx.x;
  if (i >= E_EDGES) return;
  int s = src[i], d = dst[i];
  float4 as = *(const float4*)(a_s + (size_t)s * 4);
  float4 ad = *(const float4*)(a_d + (size_t)d * 4);
  float e[4] = {as.x + ad.x, as.y + ad.y, as.z + ad.z, as.w + ad.w};
#pragma unroll
  for (int h = 0; h < 4; ++h) {
    float t = e[h]; t = t > 0.f ? t : NEG_SLOPE * t;
    atomicMax(&mkey[(size_t)d * 4 + h], f2ord(t));
  }
}

// ---------- edge pass 2: segment sum of exp ----------
__global__ void edge_sum_kernel(const int* __restrict__ src, const int* __restrict__ dst,
                                const float* __restrict__ a_s, const float* __restrict__ a_d,
                                const unsigned int* __restrict__ mkey,
                                float* __restrict__ denom) {
  int i = blockIdx.x * blockDim.x + threadIdx.x;
  if (i >= E_EDGES) return;
  int s = src[i], d = dst[i];
  float4 as = *(const float4*)(a_s + (size_t)s * 4);
  float4 ad = *(const float4*)(a_d + (size_t)d * 4);
  float e[4] = {as.x + ad.x, as.y + ad.y, as.z + ad.z, as.w + ad.w};
#pragma unroll
  for (int h = 0; h < 4; ++h) {
    float t = e[h]; t = t > 0.f ? t : NEG_SLOPE * t;
    float ex = __expf(t - ord2f(mkey[(size_t)d * 4 + h]));
    atomicAdd(&denom[(size_t)d * 4 + h], ex);
  }
}

// ---------- edge pass 3: weighted scatter of hs[src] (wave per edge, 8 ch/lane) ----------
__global__ void edge_scatter_kernel(const int* __restrict__ src, const int* __restrict__ dst,
                                    const float* __restrict__ a_s, const float* __restrict__ a_d,
                                    const unsigned int* __restrict__ mkey,
                                    const float* __restrict__ denom,
                                    const float* __restrict__ hs,
                                    float* __restrict__ out) {
  int wave = (int)((blockIdx.x * blockDim.x + threadIdx.x) >> 5);
  int lane = threadIdx.x & 31;
  if (wave >= E_EDGES) return;
  int s = src[wave], d = dst[wave];
  int h = lane >> 3;                       // 8 channels per lane -> head = lane/8
  float t = a_s[(size_t)s * 4 + h] + a_d[(size_t)d * 4 + h];
  t = t > 0.f ? t : NEG_SLOPE * t;
  float m  = ord2f(mkey[(size_t)d * 4 + h]);
  float dn = denom[(size_t)d * 4 + h];
  float alpha = __expf(t - m) / (dn + 1e-16f);
  int c0 = lane * 8;
  const float4* hp = (const float4*)(hs + (size_t)s * OUTD + c0);
  float4 h0 = hp[0], h1 = hp[1];
  float* op = out + (size_t)d * OUTD + c0;
  atomicAdd(op + 0, alpha * h0.x); atomicAdd(op + 1, alpha * h0.y);
  atomicAdd(op + 2, alpha * h0.z); atomicAdd(op + 3, alpha * h0.w);
  atomicAdd(op + 4, alpha * h1.x); atomicAdd(op + 5, alpha * h1.y);
  atomicAdd(op + 6, alpha * h1.z); atomicAdd(op + 7, alpha * h1.w);
}

// ---------- final: ReLU + LayerNorm in place (wave per row) ----------
__global__ void relu_ln_kernel(float* __restrict__ io,
                               const float* __restrict__ gamma,
                               const float* __restrict__ beta) {
  int wave = (int)((blockIdx.x * blockDim.x + threadIdx.x) >> 5);
  int lane = threadIdx.x & 31;
  if (wave >= NJ) return;
  float* row = io + (size_t)wave * OUTD;
  int c0 = lane * 8;
  float4 a = *(float4*)(row + c0);
  float4 b = *(float4*)(row + c0 + 4);
  float v[8] = {a.x, a.y, a.z, a.w, b.x, b.y, b.z, b.w};
  float s = 0.f, s2 = 0.f;
#pragma unroll
  for (int j = 0; j < 8; ++j) {
    float t = v[j] > 0.f ? v[j] : 0.f;
    v[j] = t; s += t; s2 += t * t;
  }
#pragma unroll
  for (int off = 16; off >= 1; off >>= 1) {
    s  += __shfl_xor(s,  off, 32);
    s2 += __shfl_xor(s2, off, 32);
  }
  float mu  = s  * (1.f / OUTD);
  float var = s2 * (1.f / OUTD) - mu * mu;
  float inv = rsqrtf(var + LN_EPS);
#pragma unroll
  for (int j = 0; j < 8; ++j) {
    int c = c0 + j;
    row[c] = (v[j] - mu) * inv * gamma[c] + beta[c];
  }
}

extern "C" void kernel_launch(void* const* d_in, const int* in_sizes, int n_in,
                              void* d_out, int out_size, void* d_ws, size_t ws_size,
                              hipStream_t stream) {
  (void)in_sizes; (void)n_in; (void)out_size; (void)ws_size;
  const float* x_job     = (const float*)d_in[0];
  const float* x_station = (const float*)d_in[1];
  const float* x_machine = (const float*)d_in[2];
  const float* x_robot   = (const float*)d_in[3];
  const int* srcs[5] = {(const int*)d_in[4], (const int*)d_in[6], (const int*)d_in[8],
                        (const int*)d_in[10], (const int*)d_in[12]};
  const int* dsts[5] = {(const int*)d_in[5], (const int*)d_in[7], (const int*)d_in[9],
                        (const int*)d_in[11], (const int*)d_in[13]};
  const float* Ws       = (const float*)d_in[14];
  const float* att_src  = (const float*)d_in[15];
  const float* att_dst  = (const float*)d_in[16];
  const float* biases   = (const float*)d_in[17];
  const float* W_res    = (const float*)d_in[18];
  const float* ln_gamma = (const float*)d_in[19];
  const float* ln_beta  = (const float*)d_in[20];

  const float* xs[5] = {x_station, x_station, x_machine, x_machine, x_robot};
  const int    Ns[5] = {20000, 20000, 10000, 10000, 5000};

  // ---- workspace carve-up ----
  char* ws = (char*)d_ws;
  size_t off = 0;
  auto carve = [&](size_t bytes) -> char* {
    char* p = ws + off;
    off = (off + bytes + 255) & ~(size_t)255;
    return p;
  };
  float* wfold_s = (float*)carve(5 * D_IN * HEADS * sizeof(float));
  float* wfold_d = (float*)carve(5 * D_IN * HEADS * sizeof(float));
  float* a_s[5]; for (int r = 0; r < 5; ++r) a_s[r] = (float*)carve((size_t)Ns[r] * 4 * sizeof(float));
  float* a_d[5]; for (int r = 0; r < 5; ++r) a_d[r] = (float*)carve((size_t)NJ * 4 * sizeof(float));
  unsigned int* mkey[5]; for (int r = 0; r < 5; ++r) mkey[r] = (unsigned int*)carve((size_t)NJ * 4 * sizeof(unsigned int));
  float* denom[5]; for (int r = 0; r < 5; ++r) denom[r] = (float*)carve((size_t)NJ * 4 * sizeof(float));
  float* hs[5]; for (int r = 0; r < 5; ++r) hs[r] = (float*)carve((size_t)Ns[r] * OUTD * sizeof(float));

  float* out = (float*)d_out; // doubles as the accumulator

  // 1) fold attention vectors through W
  fold_att_kernel<<<(5 * D_IN * HEADS + 255) / 256, 256, 0, stream>>>(
      Ws, att_src, att_dst, wfold_s, wfold_d);

  // 2) per-node attention scalars
  for (int r = 0; r < 5; ++r) {
    int gs = (Ns[r] * 32 + 255) / 256;
    att_proj_kernel<<<gs, 256, 0, stream>>>(xs[r], Ns[r], wfold_s + r * D_IN * HEADS, a_s[r]);
  }
  for (int r = 0; r < 5; ++r) {
    int gd = (NJ * 32 + 255) / 256;
    att_proj_kernel<<<gd, 256, 0, stream>>>(x_job, NJ, wfold_d + r * D_IN * HEADS, a_d[r]);
  }

  // 3) hs = x_src @ W_r  (fp32 WMMA)
  for (int r = 0; r < 5; ++r) {
    int ntm = (Ns[r] + 15) / 16;
    int waves = ntm * 16;
    gemm_hs_kernel<<<(waves * 32 + 255) / 256, 256, 0, stream>>>(
        xs[r], Ws + (size_t)r * D_IN * OUTD, hs[r], Ns[r]);
  }

  // 4) init softmax state
  for (int r = 0; r < 5; ++r) {
    fill_u32_kernel<<<(NJ * 4 + 255) / 256, 256, 0, stream>>>(mkey[r], 0x007FFFFFu, NJ * 4); // enc(-inf)
    fill_u32_kernel<<<(NJ * 4 + 255) / 256, 256, 0, stream>>>((unsigned int*)denom[r], 0u, NJ * 4);
  }

  // 5) init accumulator: d_out = x_job @ W_res + sum_r b_r  (fp32 WMMA)
  {
    int ntm = (NJ + 15) / 16;
    int waves = ntm * 16;
    gemm_res_init_kernel<<<(waves * 32 + 255) / 256, 256, 0, stream>>>(
        x_job, W_res, biases, out, NJ);
  }

  // 6) edge softmax passes
  int ge = (E_EDGES + 255) / 256;
  for (int r = 0; r < 5; ++r)
    edge_max_kernel<<<ge, 256, 0, stream>>>(srcs[r], dsts[r], a_s[r], a_d[r], mkey[r]);
  for (int r = 0; r < 5; ++r)
    edge_sum_kernel<<<ge, 256, 0, stream>>>(srcs[r], dsts[r], a_s[r], a_d[r], mkey[r], denom[r]);
  int gw = (E_EDGES * 32 + 255) / 256;
  for (int r = 0; r < 5; ++r)
    edge_scatter_kernel<<<gw, 256, 0, stream>>>(srcs[r], dsts[r], a_s[r], a_d[r],
                                                mkey[r], denom[r], hs[r], out);

  // 7) ReLU + LayerNorm in place
  relu_ln_kernel<<<(NJ * 32 + 255) / 256, 256, 0, stream>>>(out, ln_gamma, ln_beta);
}